// GNN_39702677684337
// MI455X (gfx1250) — compile-verified
//
#include <hip/hip_runtime.h>

#define N_NODES 50000
#define C 128
#define ROW_TILES (N_NODES / 16)   // 3125 (exact)

typedef __attribute__((ext_vector_type(2))) float v2f;
typedef __attribute__((ext_vector_type(8))) float v8f;

// ---------------- zero workspace (agg + deg contiguous) ----------------
__global__ void sage_zero_f4(float4* __restrict__ p, int n4) {
    int i = blockIdx.x * blockDim.x + threadIdx.x;
    if (i < n4) p[i] = make_float4(0.f, 0.f, 0.f, 0.f);
}

// ---------------- edge scatter: one wave32 per edge ----------------
// lane c handles features [4c, 4c+3]; all atomics resolve in L2 (agg is 25.6MB).
__global__ void sage_scatter(const float* __restrict__ xin,
                             const long long* __restrict__ src,
                             const long long* __restrict__ dst,
                             float* __restrict__ agg,
                             float* __restrict__ deg, int E) {
    const int wave = blockIdx.x * (blockDim.x >> 5) + (threadIdx.x >> 5);
    const int lane = threadIdx.x & 31;
    if (wave >= E) return;
    const long long s = src[wave];
    const long long d = dst[wave];
    const float4 v = ((const float4*)(xin + s * C))[lane];
    float* a = agg + d * C + lane * 4;
    atomicAdd(a + 0, v.x);
    atomicAdd(a + 1, v.y);
    atomicAdd(a + 2, v.z);
    atomicAdd(a + 3, v.w);
    if (lane == 0) atomicAdd(deg + d, 1.0f);
}

// ---------------- mean = agg / max(deg, 1), in place, float4 granular ----------------
__global__ void sage_normalize(float* __restrict__ agg, const float* __restrict__ deg, int n4) {
    int i = blockIdx.x * blockDim.x + threadIdx.x;   // over N*C/4
    if (i >= n4) return;
    const int n = i >> 5;                            // (i*4)/128
    const float dv = fmaxf(deg[n], 1.0f);
    float4 v = ((const float4*)agg)[i];
    v.x /= dv; v.y /= dv; v.z /= dv; v.w /= dv;
    ((float4*)agg)[i] = v;
}

// ---------------- fused fp32 WMMA GEMM: out = mean@Wl^T + bl + x@Wr^T (+ReLU) ----------
// Block = 256 threads = 8 waves. Block b owns node rows [16b, 16b+16); wave w owns
// output cols [16w, 16w+16). Each wave: 32 k-steps x 2 WMMA (Wl-term and Wr-term
// accumulate into the same v8f), bias + optional ReLU epilogue.
__global__ void __launch_bounds__(256)
sage_gemm(const float* __restrict__ mean, const float* __restrict__ xin,
          const float* __restrict__ Wl, const float* __restrict__ bl,
          const float* __restrict__ Wr, float* __restrict__ out, int relu) {
    const int lane    = threadIdx.x & 31;
    const int colTile = threadIdx.x >> 5;    // 0..7
    const int rowTile = blockIdx.x;          // 0..3124
    const int half    = lane >> 4;           // 0/1: k-group for A/B fragments
    const int l16     = lane & 15;

    const int mRow = rowTile * 16 + l16;     // A row loaded by this lane
    const int nCol = colTile * 16 + l16;     // B col / D col for this lane

    const float* mrow  = mean + (size_t)mRow * C;
    const float* xrow  = xin  + (size_t)mRow * C;
    const float* wlrow = Wl   + (size_t)nCol * C;  // B_l[k][n] = Wl[n][k]
    const float* wrrow = Wr   + (size_t)nCol * C;  // B_r[k][n] = Wr[n][k]

    v8f acc = {};
#pragma unroll
    for (int kk = 0; kk < C; kk += 4) {
        const int k0 = kk + half * 2;        // lane holds k0, k0+1 (reg 0, reg 1)
        const v2f aM = *(const v2f*)(mrow  + k0);
        const v2f aX = *(const v2f*)(xrow  + k0);
        const v2f bL = *(const v2f*)(wlrow + k0);
        const v2f bR = *(const v2f*)(wrrow + k0);
        acc = __builtin_amdgcn_wmma_f32_16x16x4_f32(false, aM, false, bL,
                                                    (short)0, acc, false, false);
        acc = __builtin_amdgcn_wmma_f32_16x16x4_f32(false, aX, false, bR,
                                                    (short)0, acc, false, false);
    }

    const float bias = bl[nCol];
    float* ocol = out + (size_t)(rowTile * 16) * C + nCol;
#pragma unroll
    for (int v = 0; v < 8; ++v) {            // D reg v -> row v + 8*half
        float val = acc[v] + bias;
        if (relu) val = fmaxf(val, 0.0f);
        ocol[(size_t)(v + 8 * half) * C] = val;
    }
}

extern "C" void kernel_launch(void* const* d_in, const int* in_sizes, int n_in,
                              void* d_out, int out_size, void* d_ws, size_t ws_size,
                              hipStream_t stream) {
    const float*     x    = (const float*)d_in[0];
    const long long* edge = (const long long*)d_in[1];   // int64 [2, E] flat
    const float*     Wl1  = (const float*)d_in[2];
    const float*     bl1  = (const float*)d_in[3];
    const float*     Wr1  = (const float*)d_in[4];
    const float*     Wl2  = (const float*)d_in[5];
    const float*     bl2  = (const float*)d_in[6];
    const float*     Wr2  = (const float*)d_in[7];

    const int E = in_sizes[1] / 2;
    const long long* src = edge;
    const long long* dst = edge + E;

    // workspace layout: [agg: N*C][deg: N][h: N*C]  (~51.4 MB)
    float* agg = (float*)d_ws;
    float* deg = agg + (size_t)N_NODES * C;
    float* h   = deg + N_NODES;
    float* outp = (float*)d_out;

    const int zeroN4 = (N_NODES * C + N_NODES) / 4;   // agg+deg contiguous
    const int normN4 = (N_NODES * C) / 4;
    dim3 zgrid((zeroN4 + 255) / 256);
    dim3 sgrid((E + 7) / 8);                          // 8 waves (edges) per block
    dim3 ngrid((normN4 + 255) / 256);
    dim3 ggrid(ROW_TILES);

    // ---- layer 1 ----
    sage_zero_f4 <<<zgrid, 256, 0, stream>>>((float4*)agg, zeroN4);
    sage_scatter <<<sgrid, 256, 0, stream>>>(x, src, dst, agg, deg, E);
    sage_normalize<<<ngrid, 256, 0, stream>>>(agg, deg, normN4);
    sage_gemm    <<<ggrid, 256, 0, stream>>>(agg, x, Wl1, bl1, Wr1, h, 1);

    // ---- layer 2 ----
    sage_zero_f4 <<<zgrid, 256, 0, stream>>>((float4*)agg, zeroN4);
    sage_scatter <<<sgrid, 256, 0, stream>>>(h, src, dst, agg, deg, E);
    sage_normalize<<<ngrid, 256, 0, stream>>>(agg, deg, normN4);
    sage_gemm    <<<ggrid, 256, 0, stream>>>(agg, h, Wl2, bl2, Wr2, outp, 0);
}